// GAT_62079457296419
// MI455X (gfx1250) — compile-verified
//
#include <hip/hip_runtime.h>
#include <hip/hip_bf16.h>

typedef __attribute__((ext_vector_type(2))) float v2f;
typedef __attribute__((ext_vector_type(8))) float v8f;

#define GAT_N_NODES 50000
#define GAT_N_EDGES 800000

// ---------------------------------------------------------------------------
// GEMM: C[n,N] = A[n,K] @ W[K,N], fp32, via V_WMMA_F32_16X16X4_F32.
// Each wave32 computes a 16 x (16*NT) output strip: one A fragment is reused
// across NT WMMAs per K-step (amortizes A loads 4x, A re-read drops from
// N/16 to N/(16*NT) passes; A fits in the 192MB L2 either way).
// Fragment layouts per ISA 7.12.2:
//   A (16x4 f32): lanes 0-15 row=lane, K={0,1} in v0,v1; lanes 16-31 K={2,3}.
//   B (4x16 f32): lanes 0-15 v0=B[0][lane], v1=B[1][lane]; lanes 16-31 K={2,3}.
//   C/D (16x16 f32): VGPR i -> row i (lanes 0-15) / row i+8 (lanes 16-31).
// n%16==0, N%(16*NT)==0, K%4==0 guaranteed by caller.
// ---------------------------------------------------------------------------
template <int NT>
__global__ void gemm_wmma_f32(const float* __restrict__ A,
                              const float* __restrict__ W,
                              float* __restrict__ C,
                              int n, int K, int N) {
  int wave    = blockIdx.x * (blockDim.x >> 5) + (threadIdx.x >> 5);
  int stripsN = N / (16 * NT);
  int strips  = (n >> 4) * stripsN;
  if (wave >= strips) return;
  int mt = wave / stripsN, ns = wave % stripsN;
  int lane = threadIdx.x & 31;
  int half = lane >> 4;     // 0: K pair {0,1}, 1: K pair {2,3}
  int l15  = lane & 15;
  int m0 = mt << 4, n0 = ns * (16 * NT);

  const float* arow = A + (size_t)(m0 + l15) * K + half * 2;
  const float* bcol = W + (size_t)(half * 2) * N + n0 + l15;

  v8f acc[NT];
#pragma unroll
  for (int t = 0; t < NT; ++t) acc[t] = (v8f){};

  for (int k0 = 0; k0 < K; k0 += 4) {
    v2f a;
    a.x = arow[k0];
    a.y = arow[k0 + 1];
    v2f b[NT];
#pragma unroll
    for (int t = 0; t < NT; ++t) {
      b[t].x = bcol[(size_t)k0 * N + t * 16];
      b[t].y = bcol[(size_t)(k0 + 1) * N + t * 16];
    }
#pragma unroll
    for (int t = 0; t < NT; ++t)
      acc[t] = __builtin_amdgcn_wmma_f32_16x16x4_f32(
          /*neg_a=*/false, a, /*neg_b=*/false, b[t],
          /*c_mod=*/(short)0, acc[t], /*reuse_a=*/false, /*reuse_b=*/false);
  }

  float* crow = C + (size_t)(m0 + (half << 3)) * N + n0 + l15;
#pragma unroll
  for (int t = 0; t < NT; ++t)
#pragma unroll
    for (int i = 0; i < 8; ++i) crow[(size_t)i * N + t * 16] = acc[t][i];
}

// el/er per (node, head); also init m = -inf, s = 0.
__global__ void attn_scores(const float* __restrict__ ft,
                            const float* __restrict__ al,
                            const float* __restrict__ ar,
                            float* __restrict__ el, float* __restrict__ er,
                            float* __restrict__ mb, float* __restrict__ sb,
                            int n, int H, int Cc) {
  int idx = blockIdx.x * blockDim.x + threadIdx.x;
  if (idx >= n * H) return;
  int node = idx / H, h = idx % H;
  const float* f = ft + ((size_t)node * H + h) * Cc;
  float sl = 0.f, sr = 0.f;
  for (int c = 0; c < Cc; ++c) {
    float v = f[c];
    sl += v * al[h * Cc + c];
    sr += v * ar[h * Cc + c];
  }
  el[idx] = sl;
  er[idx] = sr;
  mb[idx] = -__builtin_inff();
  sb[idx] = 0.f;
}

// float atomic max via sign-split (monotone int for >=0, reverse-monotone uint
// for <0); both branches are mutually consistent on the same address.
__device__ __forceinline__ void atomicMaxF(float* addr, float v) {
  if (v >= 0.f) atomicMax((int*)addr, __float_as_int(v));
  else          atomicMin((unsigned int*)addr, __float_as_uint(v));
}

__global__ void edge_logits_max(const int* __restrict__ src,
                                const int* __restrict__ dst,
                                const float* __restrict__ el,
                                const float* __restrict__ er,
                                float* __restrict__ exb,
                                float* __restrict__ mb, int E, int H) {
  int idx = blockIdx.x * blockDim.x + threadIdx.x;
  if (idx >= E * H) return;
  int e = idx / H, h = idx % H;
  int s = src[e], d = dst[e];
  float x = el[(size_t)s * H + h] + er[(size_t)d * H + h];
  x = (x > 0.f) ? x : 0.2f * x;          // leaky_relu(0.2)
  exb[idx] = x;                          // stash logit for pass 2
  atomicMaxF(&mb[(size_t)d * H + h], x);
}

__global__ void edge_exp_sum(const int* __restrict__ dst,
                             float* __restrict__ exb,
                             const float* __restrict__ mb,
                             float* __restrict__ sb, int E, int H) {
  int idx = blockIdx.x * blockDim.x + threadIdx.x;
  if (idx >= E * H) return;
  int e = idx / H, h = idx % H;
  int d = dst[e];
  float v = __expf(exb[idx] - mb[(size_t)d * H + h]);
  exb[idx] = v;
  atomicAdd(&sb[(size_t)d * H + h], v);
}

__global__ void init_bias(float* __restrict__ out, const float* __restrict__ b,
                          int total, int F) {
  int idx = blockIdx.x * blockDim.x + threadIdx.x;
  if (idx >= total) return;
  out[idx] = b[idx % F];
}

__global__ void aggregate(const int* __restrict__ src,
                          const int* __restrict__ dst,
                          const float* __restrict__ exb,
                          const float* __restrict__ sb,
                          const float* __restrict__ ft,
                          float* __restrict__ out, int E, int H, int Cc) {
  int idx = blockIdx.x * blockDim.x + threadIdx.x;
  if (idx >= E * H) return;
  int e = idx / H, h = idx % H;
  int sn = src[e], d = dst[e];
  float alpha = exb[idx] / sb[(size_t)d * H + h];
  const float* f = ft + ((size_t)sn * H + h) * Cc;
  float* o = out + ((size_t)d * H + h) * Cc;
  for (int c = 0; c < Cc; c += 4) {
    float4 v = *(const float4*)(f + c);
    atomicAdd(&o[c + 0], alpha * v.x);
    atomicAdd(&o[c + 1], alpha * v.y);
    atomicAdd(&o[c + 2], alpha * v.z);
    atomicAdd(&o[c + 3], alpha * v.w);
  }
}

__global__ void relu_k(float* __restrict__ x, int total) {
  int i = blockIdx.x * blockDim.x + threadIdx.x;
  if (i < total) x[i] = fmaxf(x[i], 0.f);
}

// relu then softmax over F (<=32) classes, one thread per node.
__global__ void relu_softmax(const float* __restrict__ in,
                             float* __restrict__ out, int n, int F) {
  int node = blockIdx.x * blockDim.x + threadIdx.x;
  if (node >= n) return;
  const float* r = in + (size_t)node * F;
  float tmp[32];
  float mx = -__builtin_inff();
  for (int c = 0; c < F; ++c) {
    float v = fmaxf(r[c], 0.f);
    tmp[c] = v;
    mx = fmaxf(mx, v);
  }
  float s = 0.f;
  for (int c = 0; c < F; ++c) {
    float v = __expf(tmp[c] - mx);
    tmp[c] = v;
    s += v;
  }
  float inv = 1.f / s;
  for (int c = 0; c < F; ++c) out[(size_t)node * F + c] = tmp[c] * inv;
}

// ---------------------------------------------------------------------------
static void run_layer(const float* in, int Din, const float* W,
                      const float* al, const float* ar, const float* b,
                      int H, int Cc, const int* src, const int* dst,
                      float* ft, float* out,
                      float* el, float* er, float* mb, float* sb, float* exb,
                      bool do_relu, hipStream_t stream) {
  const int n = GAT_N_NODES, E = GAT_N_EDGES;
  const int F = H * Cc;
  if (F % 64 == 0) {
    int strips = (n / 16) * (F / 64);
    gemm_wmma_f32<4><<<(strips + 7) / 8, 256, 0, stream>>>(in, W, ft, n, Din, F);
  } else {
    int strips = (n / 16) * (F / 32);
    gemm_wmma_f32<2><<<(strips + 7) / 8, 256, 0, stream>>>(in, W, ft, n, Din, F);
  }
  attn_scores<<<(n * H + 255) / 256, 256, 0, stream>>>(ft, al, ar, el, er, mb,
                                                       sb, n, H, Cc);
  init_bias<<<(n * F + 255) / 256, 256, 0, stream>>>(out, b, n * F, F);
  edge_logits_max<<<(E * H + 255) / 256, 256, 0, stream>>>(src, dst, el, er,
                                                           exb, mb, E, H);
  edge_exp_sum<<<(E * H + 255) / 256, 256, 0, stream>>>(dst, exb, mb, sb, E, H);
  aggregate<<<(E * H + 255) / 256, 256, 0, stream>>>(src, dst, exb, sb, ft,
                                                     out, E, H, Cc);
  if (do_relu)
    relu_k<<<(n * F + 255) / 256, 256, 0, stream>>>(out, n * F);
}

extern "C" void kernel_launch(void* const* d_in, const int* in_sizes, int n_in,
                              void* d_out, int out_size, void* d_ws,
                              size_t ws_size, hipStream_t stream) {
  const float* x   = (const float*)d_in[0];
  const int*   src = (const int*)d_in[1];
  const int*   dst = (const int*)d_in[2];
  const float* W0 = (const float*)d_in[3],  *al0 = (const float*)d_in[4],
             * ar0 = (const float*)d_in[5], *b0  = (const float*)d_in[6];
  const float* W1 = (const float*)d_in[7],  *al1 = (const float*)d_in[8],
             * ar1 = (const float*)d_in[9], *b1  = (const float*)d_in[10];
  const float* W2 = (const float*)d_in[11], *al2 = (const float*)d_in[12],
             * ar2 = (const float*)d_in[13], *b2 = (const float*)d_in[14];

  const size_t n = GAT_N_NODES;
  float* ws   = (float*)d_ws;
  float* bufP = ws;                  // n*256  (layer outputs / next input)
  float* bufQ = bufP + n * 256;      // n*256  (projected features ft)
  float* el   = bufQ + n * 256;      // n*8
  float* er   = el + n * 8;          // n*8
  float* mb   = er + n * 8;          // n*8
  float* sb   = mb + n * 8;          // n*8
  float* exb  = sb + n * 8;          // E*8

  // Layer 0: x[50000,128] -> 8 heads x 32
  run_layer(x, 128, W0, al0, ar0, b0, 8, 32, src, dst, bufQ, bufP, el, er, mb,
            sb, exb, true, stream);
  // Layer 1: [50000,256] -> 8 heads x 32
  run_layer(bufP, 256, W1, al1, ar1, b1, 8, 32, src, dst, bufQ, bufP, el, er,
            mb, sb, exb, true, stream);
  // Layer 2: [50000,256] -> 1 head x 32 (relu fused into softmax)
  run_layer(bufP, 256, W2, al2, ar2, b2, 1, 32, src, dst, bufQ, bufP, el, er,
            mb, sb, exb, false, stream);

  relu_softmax<<<(GAT_N_NODES + 255) / 256, 256, 0, stream>>>(
      bufP, (float*)d_out, GAT_N_NODES, 32);
}